// DirectPoseOutputs_5987184411030
// MI455X (gfx1250) — compile-verified
//
#include <hip/hip_runtime.h>
#include <hip/hip_bf16.h>

// Problem constants (from reference)
#define B_      16
#define CKPT    17
#define Hh      100
#define Ww      152
#define HW      (Hh * Ww)          // 15200
#define KTOP    40
#define CF      256
#define NROW    (B_ * CKPT)        // 272
#define NREG    20000
#define CT_BLOCKS ((NREG + 255) / 256)   // 79

typedef __attribute__((ext_vector_type(2))) float v2f;
typedef __attribute__((ext_vector_type(8))) float v8f;
typedef int v4i __attribute__((vector_size(16)));   // matches builtin's b128 pointee

// ---------------- CDNA5 async global->LDS helpers ----------------
// LDS byte address recovered from generic pointer low 32 bits (ISA 10.2:
// in the shared aperture these bits are the LDS address).
__device__ __forceinline__ void async_copy_b128(const float* gptr, unsigned lds_byte_addr) {
#if __has_builtin(__builtin_amdgcn_global_load_async_to_lds_b128)
    __builtin_amdgcn_global_load_async_to_lds_b128(
        (__attribute__((address_space(1))) v4i*)(unsigned long long)(const void*)gptr,
        (__attribute__((address_space(3))) v4i*)lds_byte_addr,
        0, 0);
#else
    asm volatile("global_load_async_to_lds_b128 %0, %1, off"
                 :: "v"(lds_byte_addr),
                    "v"((unsigned long long)(const void*)gptr)
                 : "memory");
#endif
}

__device__ __forceinline__ void wait_async_zero() {
#if __has_builtin(__builtin_amdgcn_s_wait_asynccnt)
    __builtin_amdgcn_s_wait_asynccnt(0);
#else
    asm volatile("s_wait_asynccnt 0" ::: "memory");
#endif
}

// ---------------- Kernel 1: ctrness partial sums (WMMA wave reduction) ----
// ctrness_i = sqrt( (min(l,r)/max(l,r)) * (min(t,b)/max(t,b)) )
// Wave sum via V_WMMA_F32_16X16X4_F32 with ones B-matrix:
//   A[m,0]=p_m, A[m,2]=p_{m+16}, A[m,1]=A[m,3]=0  =>  D[m,n] = p_m + p_{m+16}
// lane0 holds D[0..7,0], lane16 holds D[8..15,0] -> together the wave sum.
__global__ __launch_bounds__(256) void ctrness_partial_kernel(
    const float* __restrict__ rt, float* __restrict__ partials, int n)
{
    const int gid = blockIdx.x * 256 + threadIdx.x;
    float p = 0.f;
    if (gid < n) {
        const float l  = rt[gid * 4 + 0];
        const float t  = rt[gid * 4 + 1];
        const float r  = rt[gid * 4 + 2];
        const float bm = rt[gid * 4 + 3];
        const float lr = fminf(l, r) / fmaxf(l, r);
        const float tb = fminf(t, bm) / fmaxf(t, bm);
        p = sqrtf(lr * tb);
    }
    __shared__ float wsum[16];
    const int lane = threadIdx.x & 31;
    const int wv   = threadIdx.x >> 5;
#if __has_builtin(__builtin_amdgcn_wmma_f32_16x16x4_f32)
    v2f a;  a[0] = p;    a[1] = 0.f;   // K=1,3 slices zero
    v2f bo; bo[0] = 1.f; bo[1] = 1.f;  // B all ones
    v8f c = {};
    c = __builtin_amdgcn_wmma_f32_16x16x4_f32(false, a, false, bo,
                                              (short)0, c, false, false);
    const float half_sum = c[0] + c[1] + c[2] + c[3] + c[4] + c[5] + c[6] + c[7];
    if (lane == 0)  wsum[wv * 2 + 0] = half_sum;
    if (lane == 16) wsum[wv * 2 + 1] = half_sum;
#else
    float s = p;
    for (int off = 16; off > 0; off >>= 1) s += __shfl_xor(s, off, 32);
    if (lane == 0) { wsum[wv * 2 + 0] = s; wsum[wv * 2 + 1] = 0.f; }
#endif
    __syncthreads();
    if (threadIdx.x == 0) {
        float s = 0.f;
#pragma unroll
        for (int i = 0; i < 16; ++i) s += wsum[i];
        partials[blockIdx.x] = s;
    }
}

// Deterministic finalize: fixed-order sum of block partials -> mean.
__global__ void finalize_mean_kernel(const float* __restrict__ partials, int nb,
                                     float* __restrict__ mean_out)
{
    if (threadIdx.x == 0 && blockIdx.x == 0) {
        float s = 0.f;
        for (int i = 0; i < nb; ++i) s += partials[i];
        *mean_out = s * (1.0f / (float)NREG);
    }
}

// ---------------- Kernel 2: fused NMS + exact top-40 per (b,c) row -------
// One block per row. Heat plane (60.8 KB) async-staged into LDS; 3x3 NMS
// into a second LDS plane; 40 deterministic argmax rounds (lowest-index
// tiebreak, matching jax.lax.top_k ordering).
__global__ __launch_bounds__(256) void nms_topk_kernel(
    const float* __restrict__ heat, float* __restrict__ scores, int* __restrict__ inds)
{
    extern __shared__ float smem[];
    float* raw  = smem;                        // HW floats
    float* nmsb = smem + HW;                   // HW floats
    float* redV = smem + 2 * HW;               // 256 floats
    int*   redI = (int*)(smem + 2 * HW + 256); // 256 ints

    const int row = blockIdx.x;
    const int tid = threadIdx.x;
    const float* plane = heat + (size_t)row * HW;

    // Async-stage the plane into LDS, 16B granules (HW/4 = 3800, 16B aligned).
    const unsigned raw_lds = (unsigned)(unsigned long long)(const void*)raw;
    for (int i = tid; i < HW / 4; i += 256) {
        async_copy_b128(plane + i * 4, raw_lds + (unsigned)i * 16u);
    }
    wait_async_zero();
    __syncthreads();

    // 3x3 NMS with border clamping (reduce_window init = -inf => clamp is exact).
    for (int i = tid; i < HW; i += 256) {
        const int y = i / Ww;
        const int x = i - y * Ww;
        const float cv = raw[i];
        float m = cv;
        const int y0 = (y > 0) ? y - 1 : 0;
        const int y1 = (y < Hh - 1) ? y + 1 : Hh - 1;
        const int x0 = (x > 0) ? x - 1 : 0;
        const int x1 = (x < Ww - 1) ? x + 1 : Ww - 1;
        for (int yy = y0; yy <= y1; ++yy)
            for (int xx = x0; xx <= x1; ++xx)
                m = fmaxf(m, raw[yy * Ww + xx]);
        nmsb[i] = (m == cv) ? cv : 0.f;
    }
    __syncthreads();

    // Iterative exact top-K. nms values are >= 0; picked slots set to -2.
    for (int k = 0; k < KTOP; ++k) {
        float bv = -1.f;
        int   bi = 0x7fffffff;
        for (int i = tid; i < HW; i += 256) {   // ascending i + strict '>' keeps lowest index
            const float v = nmsb[i];
            if (v > bv) { bv = v; bi = i; }
        }
        redV[tid] = bv; redI[tid] = bi;
        __syncthreads();
        for (int s = 128; s > 0; s >>= 1) {
            if (tid < s) {
                const float ov = redV[tid + s]; const int oi = redI[tid + s];
                if (ov > redV[tid] || (ov == redV[tid] && oi < redI[tid])) {
                    redV[tid] = ov; redI[tid] = oi;
                }
            }
            __syncthreads();
        }
        if (tid == 0) {
            scores[row * KTOP + k] = redV[0];
            inds[row * KTOP + k]   = redI[0];
            nmsb[redI[0]] = -2.f;
        }
        __syncthreads();
    }
}

// ---------------- Kernel 3: gather + scale + bias ------------------------
// out[b, j, c] = feat[b, c, ind(b,j)] * score(b,j) + mean
__global__ __launch_bounds__(256) void gather_scale_kernel(
    const float* __restrict__ feat, const float* __restrict__ scores,
    const int* __restrict__ inds, const float* __restrict__ mean,
    float* __restrict__ out)
{
    const int j = blockIdx.x;                 // 0..679  (cls*40 + k)
    const int b = blockIdx.y;                 // 0..15
    const int c = threadIdx.x;                // 0..255
    const int rowj = b * (CKPT * KTOP) + j;
    const int ind  = inds[rowj];
    const float s  = scores[rowj];
    const float m  = mean[0];
    const float v  = feat[((size_t)b * CF + c) * HW + ind];
    out[(size_t)rowj * CF + c] = fmaf(v, s, m);
}

// ---------------- Launch --------------------------------------------------
extern "C" void kernel_launch(void* const* d_in, const int* in_sizes, int n_in,
                              void* d_out, int out_size, void* d_ws, size_t ws_size,
                              hipStream_t stream)
{
    const float* d_heat = (const float*)d_in[0];   // (16,17,100,152)
    const float* d_feat = (const float*)d_in[1];   // (16,256,100,152)
    const float* d_rt   = (const float*)d_in[2];   // (20000,4)
    float* out = (float*)d_out;                    // (16,680,256)

    char*  ws       = (char*)d_ws;
    float* mean     = (float*)(ws + 0);
    float* partials = (float*)(ws + 256);                    // CT_BLOCKS floats
    float* scores   = (float*)(ws + 256 + 1024);             // NROW*KTOP floats
    int*   inds     = (int*)  (ws + 256 + 1024 + NROW * KTOP * sizeof(float));

    ctrness_partial_kernel<<<CT_BLOCKS, 256, 0, stream>>>(d_rt, partials, NREG);
    finalize_mean_kernel<<<1, 32, 0, stream>>>(partials, CT_BLOCKS, mean);

    const size_t lds_bytes = (size_t)(2 * HW + 256) * sizeof(float) + 256 * sizeof(int);
    (void)hipFuncSetAttribute((const void*)nms_topk_kernel,
                              hipFuncAttributeMaxDynamicSharedMemorySize,
                              (int)lds_bytes);
    nms_topk_kernel<<<NROW, 256, lds_bytes, stream>>>(d_heat, scores, inds);

    dim3 g3(CKPT * KTOP, B_);
    gather_scale_kernel<<<g3, CF, 0, stream>>>(d_feat, scores, inds, mean, out);
}